// ConvQuadInterp3d_2353642078938
// MI455X (gfx1250) — compile-verified
//
#include <hip/hip_runtime.h>
#include <math.h>

// Problem constants (from reference setup_inputs)
#define BB 2
#define CC 4
#define DDIM 8
#define HDIM 512
#define WDIM 512
#define HT 16            // tile height per block
#define WT 32            // tile width per block
#define HHALO 2
#define WHALO 4          // 4 (not 2) so tile rows start 16B-aligned in global mem
#define LH (HT + 2 * HHALO)          // 20
#define LW (WT + 2 * WHALO)          // 40 (10 x b128 chunks per row)
#define LWC (LW / 4)                 // 10 chunks
#define TILE_N (DDIM * LH * LW)      // 6400 floats = 25.6 KB LDS
#define NCHUNK (DDIM * LH * LWC)     // 1600 b128 chunks
#define NTHREADS (WT * HT)           // 512 threads = 16 wave32 waves
#define NITERS 5
#define KBONUS 10.0f
#define KMAXSH 0.6f
#define KEPS 1e-7f

__device__ __forceinline__ float tget(const float* t, int d, int hl, int wl) {
    return t[(d * LH + hl) * LW + wl];
}

// CDNA5 async global->LDS copies (tracked by ASYNCcnt).
// lds_off: byte offset within the workgroup LDS allocation (low 32 bits of a
// generic shared pointer). gaddr: 64-bit global address (16B-aligned for b128).
__device__ __forceinline__ void async_ld_b128(unsigned lds_off, unsigned long long gaddr) {
    asm volatile("global_load_async_to_lds_b128 %0, %1, off"
                 :
                 : "v"(lds_off), "v"(gaddr)
                 : "memory");
}

__device__ __forceinline__ void wait_async0() {
    asm volatile("s_wait_asynccnt 0" ::: "memory");
}

__global__ __launch_bounds__(NTHREADS) void quadinterp3d_kernel(
        const float* __restrict__ x, float* __restrict__ out) {
    __shared__ __align__(16) float tile[TILE_N];

    const int tx = threadIdx.x;             // 0..31 -> w
    const int ty = threadIdx.y;             // 0..15 -> h
    const int tid = ty * WT + tx;           // 0..511
    const int w0 = blockIdx.x * WT;
    const int h0 = blockIdx.y * HT;
    const int bc = blockIdx.z;              // 0..7 (B*C)

    const size_t DHW = (size_t)DDIM * HDIM * WDIM;
    const float* xb = x + (size_t)bc * DHW;

    // LDS byte offset of tile[0]: low 32 bits of the generic shared pointer.
    const unsigned lds_base = (unsigned)(unsigned long long)(uintptr_t)(&tile[0]);

    // ---- Stage halo tile into LDS: one 16B async chunk per iteration ----
    // Chunks are 4-float aligned in both LDS and global W (W=512 is a multiple
    // of 4), so every chunk is either fully inside the image or fully outside.
    const float4 ninf4 = make_float4(-__builtin_inff(), -__builtin_inff(),
                                     -__builtin_inff(), -__builtin_inff());
    for (int i = tid; i < NCHUNK; i += NTHREADS) {
        int d  = i / (LH * LWC);
        int r  = i - d * (LH * LWC);
        int hl = r / LWC;
        int c4 = r - hl * LWC;
        int gh = h0 - HHALO + hl;
        int gw = w0 - WHALO + 4 * c4;
        int lidx = (d * LH + hl) * LW + 4 * c4;
        if ((unsigned)gh < (unsigned)HDIM && (unsigned)gw < (unsigned)WDIM) {
            const float* gp = xb + ((size_t)d * HDIM + gh) * WDIM + gw;
            async_ld_b128(lds_base + 4u * (unsigned)lidx,
                          (unsigned long long)(uintptr_t)gp);
        } else {
            // Out-of-image halo: -inf, matching reference's -inf NMS padding.
            *(float4*)(&tile[lidx]) = ninf4;
        }
    }
    wait_async0();        // this wave's async transfers have landed in LDS
    __syncthreads();      // all waves' tile writes visible

    const int hc = ty + HHALO;  // local center row
    const int wc = tx + WHALO;  // local center col
    const int gh = h0 + ty;
    const int gw = w0 + tx;

    // Per-plane 3x3 maxima for this (h,w) column: reused by 3 consecutive d's.
    float m3[DDIM];
#pragma unroll
    for (int d = 0; d < DDIM; ++d) {
        float m = tget(tile, d, hc - 1, wc - 1);
        m = fmaxf(m, tget(tile, d, hc - 1, wc));
        m = fmaxf(m, tget(tile, d, hc - 1, wc + 1));
        m = fmaxf(m, tget(tile, d, hc,     wc - 1));
        m = fmaxf(m, tget(tile, d, hc,     wc));
        m = fmaxf(m, tget(tile, d, hc,     wc + 1));
        m = fmaxf(m, tget(tile, d, hc + 1, wc - 1));
        m = fmaxf(m, tget(tile, d, hc + 1, wc));
        m = fmaxf(m, tget(tile, d, hc + 1, wc + 1));
        m3[d] = m;
    }

    const size_t zb    = (size_t)bc * 3 * DHW;                        // coords base
    const size_t ybase = (size_t)BB * CC * 3 * DHW + (size_t)bc * DHW; // y_max base

#pragma unroll 1
    for (int d = 0; d < DDIM; ++d) {
        const float c = tget(tile, d, hc, wc);
        float mx = m3[d];
        if (d > 0)        mx = fmaxf(mx, m3[d - 1]);
        if (d < DDIM - 1) mx = fmaxf(mx, m3[d + 1]);
        const bool nms = (c == mx);

        // Defaults (non-maximum / invalid): integer grid coords, y = x (+bonus if nms)
        float cs = (float)d, cx = (float)gw, cy = (float)gh;
        float yv = c + (nms ? KBONUS : 0.0f);

        if (nms) {
            int cd = d, ch = gh, cw = gw;
            float shx = 0.f, shy = 0.f, shs = 0.f, gds = 0.f;
            bool valid = true;
#pragma unroll 1
            for (int it = 0; it < NITERS; ++it) {
                const int di = min(max(cd, 1), DDIM - 2);
                const int hi = min(max(ch, 1), HDIM - 2);
                const int wi = min(max(cw, 1), WDIM - 2);
                const int hl = hi - h0 + HHALO;   // in [ty+1, ty+3]
                const int wl = wi - w0 + WHALO;   // in [tx+3, tx+5]

                float v[3][3][3];
#pragma unroll
                for (int a = 0; a < 3; ++a)
#pragma unroll
                    for (int b = 0; b < 3; ++b)
#pragma unroll
                        for (int e = 0; e < 3; ++e)
                            v[a][b][e] = tget(tile, di - 1 + a, hl - 1 + b, wl - 1 + e);

                const float ctr = v[1][1][1];
                const float gx = 0.5f * (v[1][1][2] - v[1][1][0]);
                const float gy = 0.5f * (v[1][2][1] - v[1][0][1]);
                const float gs = 0.5f * (v[2][1][1] - v[0][1][1]);
                const float dxx = v[1][1][2] - 2.0f * ctr + v[1][1][0];
                const float dyy = v[1][2][1] - 2.0f * ctr + v[1][0][1];
                const float dss = v[2][1][1] - 2.0f * ctr + v[0][1][1];
                const float dxy = 0.25f * (v[1][2][2] - v[1][2][0] - v[1][0][2] + v[1][0][0]);
                const float dxs = 0.25f * (v[2][1][2] - v[2][1][0] - v[0][1][2] + v[0][1][0]);
                const float dys = 0.25f * (v[2][2][1] - v[2][0][1] - v[0][2][1] + v[0][0][1]);

                const float r0 = -gx, r1 = -gy, r2 = -gs;
                const float cf00 = dyy * dss - dys * dys;
                const float cf01 = dxy * dss - dys * dxs;
                const float cf02 = dxy * dys - dyy * dxs;
                const float det = dxx * cf00 - dxy * cf01 + dxs * cf02;
                if (!(fabsf(det) > KEPS)) { valid = false; break; }

                const float t0 = r1 * dss - dys * r2;
                const float t1 = r1 * dys - dyy * r2;
                const float t2 = dxy * r2 - r1 * dxs;
                const float t3 = dyy * r2 - r1 * dys;
                shx = (r0 * cf00 - dxy * t0 + dxs * t1) / det;
                shy = (dxx * t0 - r0 * cf01 + dxs * t2) / det;
                shs = (dxx * t3 - dxy * t2 + r0 * cf02) / det;
                gds = gx * shx + gy * shy + gs * shs;

                cw += (shx > KMAXSH) ? 1 : ((shx < -KMAXSH) ? -1 : 0);
                ch += (shy > KMAXSH) ? 1 : ((shy < -KMAXSH) ? -1 : 0);
                cd += (shs > KMAXSH) ? 1 : ((shs < -KMAXSH) ? -1 : 0);

                if (abs(cd - d) > 1 || abs(ch - gh) > 1 || abs(cw - gw) > 1) {
                    valid = false; break;
                }
            }
            if (valid) {
                cs = (float)cd + shs;
                cx = (float)cw + shx;
                cy = (float)ch + shy;
                yv = c + 0.5f * gds + KBONUS;
            }
        }

        // Write-once streaming outputs: non-temporal so they don't evict x
        // (64 MB, reused across block halos) from the 192 MB L2.
        const size_t vox = ((size_t)d * HDIM + gh) * WDIM + gw;
        __builtin_nontemporal_store(cs, &out[zb + 0 * DHW + vox]);  // coord_s
        __builtin_nontemporal_store(cx, &out[zb + 1 * DHW + vox]);  // coord_x
        __builtin_nontemporal_store(cy, &out[zb + 2 * DHW + vox]);  // coord_y
        __builtin_nontemporal_store(yv, &out[ybase + vox]);         // y_max
    }
}

extern "C" void kernel_launch(void* const* d_in, const int* in_sizes, int n_in,
                              void* d_out, int out_size, void* d_ws, size_t ws_size,
                              hipStream_t stream) {
    (void)in_sizes; (void)n_in; (void)out_size; (void)d_ws; (void)ws_size;
    const float* x = (const float*)d_in[0];
    float* out = (float*)d_out;

    dim3 grid(WDIM / WT, HDIM / HT, BB * CC);  // 16 x 32 x 8 = 4096 blocks
    dim3 block(WT, HT);                         // 32 x 16 = 512 threads (16 waves)
    quadinterp3d_kernel<<<grid, block, 0, stream>>>(x, out);
}